// LSTM_38680475468350
// MI455X (gfx1250) — compile-verified
//
#include <hip/hip_runtime.h>
#include <stdint.h>

// LSTM: B=4096 seqs, T=2048 steps, F=1, H=3. One lane per sequence.
// - X staged via CDNA5 Tensor Data Mover (double buffered, HW row padding
//   -> stride-33 LDS rows, bank-conflict free), one DMA per single-wave WG.
// - Single-wave workgroups: barrier ops are S_NOP per ISA; s_wait_tensorcnt
//   alone orders DMA -> LDS reads within the wave.
// - 12 gate pre-activations packed as 6x float2 -> v_pk_fma_f32 (VOP3P).
// - All per-step tanhs issued as batched back-to-back V_TANH_F32: each TRANS
//   op is the "1 independent op" hazard cover for its predecessor, so only
//   one v_nop per batch instead of one per tanh.
// - sigmoid's x/2 scale pre-folded into i/f/o weights: sig = fma(.5,tanh,.5).
// WMMA at M16/N12/K3 would be <6% utilized + per-step cross-lane transposes.

#define HH        3
#define GATES     12
#define T_SEQ     2048
#define BB        32      // threads per block = one wave (single-wave WG)
#define TS        32      // timesteps per staged tile
#define ROWSTRIDE 33      // 32 payload + 1 pad dword (TDM pad_interval/amount)

typedef unsigned int u32x4 __attribute__((ext_vector_type(4)));
typedef int          i32x4 __attribute__((ext_vector_type(4)));
typedef int          i32x8 __attribute__((ext_vector_type(8)));
typedef float        v2f   __attribute__((ext_vector_type(2)));

#if __has_builtin(__builtin_amdgcn_tensor_load_to_lds)
#define HAVE_TDM 1
#else
#define HAVE_TDM 0
#endif

#if __has_builtin(__builtin_amdgcn_tanhf)
#define TANH1(x) __builtin_amdgcn_tanhf(x)
#elif __has_builtin(__builtin_amdgcn_tanh_f32)
#define TANH1(x) __builtin_amdgcn_tanh_f32(x)
#endif

__device__ __forceinline__ float fast_exp2(float x) {
#if __has_builtin(__builtin_amdgcn_exp2f)
  return __builtin_amdgcn_exp2f(x);
#else
  return exp2f(x);
#endif
}
__device__ __forceinline__ float fast_rcp(float x) {
#if __has_builtin(__builtin_amdgcn_rcpf)
  return __builtin_amdgcn_rcpf(x);
#else
  return 1.0f / x;
#endif
}
__device__ __forceinline__ float tanh_sw(float x) {
  float e = fast_exp2(2.88539008177792681472f * x);
  return fmaf(-2.0f, fast_rcp(e + 1.0f), 1.0f);
}

// 12 independent tanhs: one TRANS clause, single trailing v_nop.
__device__ __forceinline__ void tanh_batch12(const float* __restrict__ g,
                                             float* __restrict__ t) {
#ifdef TANH1
#pragma unroll
  for (int k = 0; k < GATES; ++k) t[k] = TANH1(g[k]);
#elif defined(__HIP_DEVICE_COMPILE__)
  asm("v_tanh_f32 %0, %12\n\t"
      "v_tanh_f32 %1, %13\n\t"
      "v_tanh_f32 %2, %14\n\t"
      "v_tanh_f32 %3, %15\n\t"
      "v_tanh_f32 %4, %16\n\t"
      "v_tanh_f32 %5, %17\n\t"
      "v_tanh_f32 %6, %18\n\t"
      "v_tanh_f32 %7, %19\n\t"
      "v_tanh_f32 %8, %20\n\t"
      "v_tanh_f32 %9, %21\n\t"
      "v_tanh_f32 %10, %22\n\t"
      "v_tanh_f32 %11, %23\n\t"
      "v_nop"
      : "=&v"(t[0]), "=&v"(t[1]), "=&v"(t[2]), "=&v"(t[3]),
        "=&v"(t[4]), "=&v"(t[5]), "=&v"(t[6]), "=&v"(t[7]),
        "=&v"(t[8]), "=&v"(t[9]), "=&v"(t[10]), "=&v"(t[11])
      : "v"(g[0]), "v"(g[1]), "v"(g[2]), "v"(g[3]),
        "v"(g[4]), "v"(g[5]), "v"(g[6]), "v"(g[7]),
        "v"(g[8]), "v"(g[9]), "v"(g[10]), "v"(g[11]));
#else
  for (int k = 0; k < GATES; ++k) t[k] = tanh_sw(g[k]);
#endif
}

// 3 independent tanhs (cell states): one TRANS clause, one v_nop.
__device__ __forceinline__ void tanh_batch3(const float* __restrict__ c,
                                            float* __restrict__ t) {
#ifdef TANH1
#pragma unroll
  for (int k = 0; k < HH; ++k) t[k] = TANH1(c[k]);
#elif defined(__HIP_DEVICE_COMPILE__)
  asm("v_tanh_f32 %0, %3\n\t"
      "v_tanh_f32 %1, %4\n\t"
      "v_tanh_f32 %2, %5\n\t"
      "v_nop"
      : "=&v"(t[0]), "=&v"(t[1]), "=&v"(t[2])
      : "v"(c[0]), "v"(c[1]), "v"(c[2]));
#else
  for (int k = 0; k < HH; ++k) t[k] = tanh_sw(c[k]);
#endif
}

__device__ __forceinline__ void wait_tensorcnt0() {
#if __has_builtin(__builtin_amdgcn_s_wait_tensorcnt)
  __builtin_amdgcn_s_wait_tensorcnt(0);
#elif defined(__HIP_DEVICE_COMPILE__) && HAVE_TDM
  asm volatile("s_wait_tensorcnt 0" ::: "memory");
#endif
}

// TDM 2D tile load: BB rows x TS cols f32, row stride T_SEQ elems, into LDS
// at lds_base with +1 dword pad per 32 dwords (row stride 33 floats).
__device__ __forceinline__ void tdm_load_tile(uint32_t lds_base, uint64_t gaddr) {
#if HAVE_TDM
  u32x4 g0;
  g0[0] = 1u;                                   // count=1 (valid user descriptor)
  g0[1] = lds_base;                             // LDS byte address
  g0[2] = (uint32_t)gaddr;                      // global_addr[31:0]
  g0[3] = (uint32_t)((gaddr >> 32) & 0x01FFFFFFu) | (2u << 30); // addr[56:32], type=2

  i32x8 g1;
  g1[0] = (2 << 16)      // data_size = 4 bytes
        | (1 << 20)      // pad_enable
        | (4 << 22);     // pad_interval: 2^(4+1)=32 dwords; pad_amount=0 -> 1 dword
  g1[1] = (T_SEQ & 0xFFFF) << 16;               // tensor_dim0[15:0]   @ bits 63:48
  g1[2] = (4096 & 0xFFFF) << 16;                // dim0 hi=0; tensor_dim1[15:0] @ 95:80
  g1[3] = (TS << 16);                           // dim1 hi=0; tile_dim0 @ 127:112
  g1[4] = BB;                                   // tile_dim1 @ 143:128; tile_dim2=0
  g1[5] = T_SEQ;                                // tensor_dim0_stride[31:0]
  g1[6] = 0;                                    // stride0 hi; tensor_dim1_stride lo
  g1[7] = 0;                                    // tensor_dim1_stride hi (unused for 2D)

  i32x4 gz = {0, 0, 0, 0};
#if defined(__clang_major__) && (__clang_major__ >= 23)
  i32x8 gz8 = {0, 0, 0, 0, 0, 0, 0, 0};
  __builtin_amdgcn_tensor_load_to_lds(g0, g1, gz, gz, gz8, 0);
#else
  __builtin_amdgcn_tensor_load_to_lds(g0, g1, gz, gz, 0);
#endif
#else
  (void)lds_base; (void)gaddr;
#endif
}

// 32 recurrence steps from one padded LDS row. Gates packed as 6x float2.
__device__ __forceinline__ void lstm_steps(const float* __restrict__ xr,
                                           const v2f* __restrict__ wihp,
                                           const v2f* __restrict__ bsp,
                                           const v2f* __restrict__ w0p,
                                           const v2f* __restrict__ w1p,
                                           const v2f* __restrict__ w2p,
                                           float* __restrict__ h,
                                           float* __restrict__ c) {
#pragma unroll 4
  for (int j = 0; j < TS; ++j) {
    float xv = xr[j];
    v2f xv2 = {xv, xv};
    v2f h0 = {h[0], h[0]}, h1 = {h[1], h[1]}, h2 = {h[2], h[2]};
    float g[GATES];
#pragma unroll
    for (int p = 0; p < GATES / 2; ++p) {
      v2f a = __builtin_elementwise_fma(xv2, wihp[p], bsp[p]);
      a = __builtin_elementwise_fma(h0, w0p[p], a);
      a = __builtin_elementwise_fma(h1, w1p[p], a);
      a = __builtin_elementwise_fma(h2, w2p[p], a);
      g[2 * p]     = a.x;
      g[2 * p + 1] = a.y;
    }
    // 12 gate tanhs in one hazard-amortized TRANS batch.
    float t[GATES];
    tanh_batch12(g, t);
    float og[HH];
#pragma unroll
    for (int k = 0; k < HH; ++k) {
      float ig = fmaf(0.5f, t[k], 0.5f);           // sigmoid(i) (half-scaled)
      float fg = fmaf(0.5f, t[HH + k], 0.5f);      // sigmoid(f)
      og[k]    = fmaf(0.5f, t[3 * HH + k], 0.5f);  // sigmoid(o)
      c[k] = fmaf(fg, c[k], ig * t[2 * HH + k]);   // t[6+k] = tanh(g) full-scale
    }
    float tc[HH];
    tanh_batch3(c, tc);
#pragma unroll
    for (int k = 0; k < HH; ++k) h[k] = og[k] * tc[k];
  }
}

__global__ __launch_bounds__(BB) void lstm_h3_kernel(
    const float* __restrict__ X, const float* __restrict__ Wih,
    const float* __restrict__ Whh, const float* __restrict__ bih,
    const float* __restrict__ bhh, float* __restrict__ out) {
  __shared__ float tile[2][BB * ROWSTRIDE];

  const int tid = threadIdx.x;
  const int b0  = blockIdx.x * BB;

  // Tiny weights -> packed register pairs; sigmoid gates (i,f,o) pre-scaled
  // by 0.5 so sigmoid needs no input multiply. Gate k=6..8 (g) stays full.
  v2f wihp[GATES / 2], bsp[GATES / 2];
  v2f w0p[GATES / 2], w1p[GATES / 2], w2p[GATES / 2];
#pragma unroll
  for (int p = 0; p < GATES / 2; ++p) {
    const int k0 = 2 * p, k1 = 2 * p + 1;
    const float s0 = (k0 >= 2 * HH && k0 < 3 * HH) ? 1.0f : 0.5f;
    const float s1 = (k1 >= 2 * HH && k1 < 3 * HH) ? 1.0f : 0.5f;
    wihp[p] = (v2f){s0 * Wih[k0], s1 * Wih[k1]};
    bsp[p]  = (v2f){s0 * (bih[k0] + bhh[k0]), s1 * (bih[k1] + bhh[k1])};
    w0p[p]  = (v2f){s0 * Whh[k0 * HH + 0], s1 * Whh[k1 * HH + 0]};
    w1p[p]  = (v2f){s0 * Whh[k0 * HH + 1], s1 * Whh[k1 * HH + 1]};
    w2p[p]  = (v2f){s0 * Whh[k0 * HH + 2], s1 * Whh[k1 * HH + 2]};
  }

  float h[HH] = {0.f, 0.f, 0.f};
  float c[HH] = {0.f, 0.f, 0.f};

#if HAVE_TDM
  const uint64_t xbase = (uint64_t)(uintptr_t)X + (uint64_t)b0 * T_SEQ * 4ull;
  uint32_t ldsb0 = (uint32_t)(uintptr_t)(&tile[0][0]);
  uint32_t ldsb1 = (uint32_t)(uintptr_t)(&tile[1][0]);

  tdm_load_tile(ldsb0, xbase);   // whole block is one wave: it drives its DMA

  int buf = 0;
  for (int t0 = 0; t0 < T_SEQ; t0 += TS) {
    wait_tensorcnt0();           // in-flight tile landed (intra-wave ordering)
    __syncthreads();             // single-wave WG: S_NOP per ISA; keeps LDS order
    if (t0 + TS < T_SEQ)
      tdm_load_tile(buf ? ldsb0 : ldsb1,
                    xbase + (uint64_t)(t0 + TS) * 4ull);   // overlap next DMA
    lstm_steps(&tile[buf][tid * ROWSTRIDE], wihp, bsp, w0p, w1p, w2p, h, c);
    buf ^= 1;
  }
#else
  // Fallback: cooperative coalesced staging (compile safety net).
  for (int t0 = 0; t0 < T_SEQ; t0 += TS) {
    __syncthreads();
#pragma unroll
    for (int i = 0; i < TS; ++i) {
      int e = i * BB + tid;
      int row = e >> 5, col = e & 31;
      tile[0][row * ROWSTRIDE + col] =
          X[(uint64_t)(b0 + row) * T_SEQ + t0 + col];
    }
    __syncthreads();
    lstm_steps(&tile[0][tid * ROWSTRIDE], wihp, bsp, w0p, w1p, w2p, h, c);
  }
#endif

  const int b = b0 + tid;
#pragma unroll
  for (int k = 0; k < HH; ++k) out[b * HH + k] = h[k];
}

extern "C" void kernel_launch(void* const* d_in, const int* in_sizes, int n_in,
                              void* d_out, int out_size, void* d_ws, size_t ws_size,
                              hipStream_t stream) {
  const float* X   = (const float*)d_in[0];
  const float* Wih = (const float*)d_in[1];
  const float* Whh = (const float*)d_in[2];
  const float* bih = (const float*)d_in[3];
  const float* bhh = (const float*)d_in[4];
  float* out = (float*)d_out;

  int B    = in_sizes[0] / T_SEQ;      // F == 1
  int grid = (B + BB - 1) / BB;        // 4096/32 = 128 single-wave blocks
  lstm_h3_kernel<<<grid, BB, 0, stream>>>(X, Wih, Whh, bih, bhh, out);
  (void)n_in; (void)out_size; (void)d_ws; (void)ws_size;
}